// ISWTReconstruction_2551210574400
// MI455X (gfx1250) — compile-verified
//
#include <hip/hip_runtime.h>

// ISWT db4, 3 levels, T=4096, rows = B*N = 4096.
// Memory-bound (335 MB @ 23.3 TB/s ~ 14us). One workgroup (8 wave32) per row:
// stage the interleaved (t, band) row into LDS with gfx1250 async
// global->LDS B128 copies (ASYNCcnt path), run the 3 dilated synthesis levels
// out of LDS, stream results back with non-temporal coalesced b32 stores.

#define T_LEN   4096
#define T_MASK  4095
#define NT      256            // 8 wave32 per workgroup
#define EPT     (T_LEN / NT)   // 16 elements per thread per phase

#if defined(__has_builtin)
#if __has_builtin(__builtin_amdgcn_global_load_async_to_lds_b128)
#define HAVE_ASYNC_LDS 1
#endif
#endif

// The builtin expects vector-of-4-int pointers in AS(1)/AS(3).
typedef int v4i __attribute__((vector_size(16)));
typedef __attribute__((address_space(1))) v4i g_v4i;   // global
typedef __attribute__((address_space(3))) v4i l_v4i;   // LDS

static __device__ __forceinline__ void wait_async0() {
#if defined(__has_builtin) && __has_builtin(__builtin_amdgcn_s_wait_asynccnt)
    __builtin_amdgcn_s_wait_asynccnt(0);
#else
    asm volatile("s_wait_asynccnt 0" ::: "memory");
#endif
}

__global__ __launch_bounds__(NT)
void iswt_db4_l3_kernel(const float4* __restrict__ coeffs,
                        float* __restrict__ out)
{
    // dec_lo filter h[k]; WL[k] = h[k], WH[k] = -(-1)^k * h[7-k].
    // constexpr locals -> folded into VALU inline literals (no SMEM fetch).
    constexpr float WL[8] = {
        -0.010597401784997278f,  0.032883011666982945f,
         0.030841381835986965f, -0.18703481171888114f,
        -0.02798376941698385f,   0.6308807679295904f,
         0.7148465705525415f,    0.23037781330885523f };
    constexpr float WH[8] = {
        -0.23037781330885523f,   0.7148465705525415f,
        -0.6308807679295904f,   -0.02798376941698385f,
         0.18703481171888114f,   0.030841381835986965f,
        -0.032883011666982945f, -0.010597401784997278f };

    // Dynamic LDS layout:
    //   raw  : float4[4096]  (64 KB)  interleaved bands {cD1, cD2, cD3, cA3} (.x..w)
    //   tmp1 : float [4096]  (16 KB)  level-1 output
    //   tmp2 : float [4096]  (16 KB)  level-2 output
    extern __shared__ __align__(16) unsigned char smem[];
    float4* raw  = reinterpret_cast<float4*>(smem);
    float*  tmp1 = reinterpret_cast<float*>(smem + T_LEN * sizeof(float4));
    float*  tmp2 = reinterpret_cast<float*>(smem + T_LEN * sizeof(float4)
                                                 + T_LEN * sizeof(float));

    const int tid = threadIdx.x;
    const int row = blockIdx.x;
    const float4* __restrict__ src = coeffs + (size_t)row * T_LEN;

    // ---- Stage the full row into LDS (coalesced b128 per time-step) ----
#ifdef HAVE_ASYNC_LDS
    #pragma unroll
    for (int i = 0; i < EPT; ++i) {
        const int t = tid + i * NT;
        __builtin_amdgcn_global_load_async_to_lds_b128(
            (g_v4i*)(src + t), (l_v4i*)(raw + t), 0, 0);
    }
    wait_async0();
#else
    #pragma unroll
    for (int i = 0; i < EPT; ++i) {
        const int t = tid + i * NT;
        raw[t] = src[t];
    }
#endif
    __syncthreads();

    // ---- Level 1: dilation 4, lo = band3 (.w), hi = band2 (.z) ----
    #pragma unroll 4
    for (int i = 0; i < EPT; ++i) {
        const int t = tid + i * NT;
        float acc = 0.0f;
        #pragma unroll
        for (int k = 0; k < 8; ++k) {
            const int idx = (t + (k - 4) * 4) & T_MASK;
            const float4 v = raw[idx];          // ds_load_b64 of (.z,.w) pair
            acc = fmaf(v.w, WL[k], acc);
            acc = fmaf(v.z, WH[k], acc);
        }
        tmp1[t] = 0.5f * acc;
    }
    __syncthreads();

    // ---- Level 2: dilation 2, lo = tmp1, hi = band1 (.y) ----
    #pragma unroll 4
    for (int i = 0; i < EPT; ++i) {
        const int t = tid + i * NT;
        float acc = 0.0f;
        #pragma unroll
        for (int k = 0; k < 8; ++k) {
            const int idx = (t + (k - 4) * 2) & T_MASK;
            acc = fmaf(tmp1[idx],   WL[k], acc);
            acc = fmaf(raw[idx].y,  WH[k], acc);
        }
        tmp2[t] = 0.5f * acc;
    }
    __syncthreads();

    // ---- Level 3: dilation 1, lo = tmp2, hi = band0 (.x) -> global ----
    float* __restrict__ dst = out + (size_t)row * T_LEN;
    #pragma unroll 4
    for (int i = 0; i < EPT; ++i) {
        const int t = tid + i * NT;
        float acc = 0.0f;
        #pragma unroll
        for (int k = 0; k < 8; ++k) {
            const int idx = (t + (k - 4)) & T_MASK;
            acc = fmaf(tmp2[idx],   WL[k], acc);
            acc = fmaf(raw[idx].x,  WH[k], acc);
        }
        // write-once stream: non-temporal store keeps L2 for the input stream
        __builtin_nontemporal_store(0.5f * acc, dst + t);
    }
}

extern "C" void kernel_launch(void* const* d_in, const int* in_sizes, int n_in,
                              void* d_out, int out_size, void* d_ws, size_t ws_size,
                              hipStream_t stream)
{
    (void)n_in; (void)d_ws; (void)ws_size; (void)out_size;
    const float4* coeffs = (const float4*)d_in[0];
    float* out = (float*)d_out;

    const int rows = in_sizes[0] / (T_LEN * 4);   // B*N = 4096
    const size_t lds_bytes = T_LEN * sizeof(float4) + 2 * T_LEN * sizeof(float); // 96 KB

    hipLaunchKernelGGL(iswt_db4_l3_kernel,
                       dim3(rows), dim3(NT), lds_bytes, stream,
                       coeffs, out);
}